// Sinkhorn_10050223473362
// MI455X (gfx1250) — compile-verified
//
#include <hip/hip_runtime.h>
#include <math.h>

typedef __attribute__((ext_vector_type(2))) float v2f;
typedef __attribute__((ext_vector_type(8))) float v8f;

#define INV_LAMBDA 20.0f          // 1 / 0.05
#define NB   8                    // batches
#define NN   2048                 // matrix dim
#define NIT  10                   // Sinkhorn iterations
#define RCHUNKS 16                // row chunks for column pass
#define CHUNK_ROWS (NN / RCHUNKS) // 128
#define CGROUPS 4                 // column groups of 512 (8 waves x 64 cols)

// ---------------------------------------------------------------- init C = 0
__global__ __launch_bounds__(256) void sk_init_zero(float* __restrict__ p, int n) {
    int i = blockIdx.x * 256 + threadIdx.x;
    if (i < n) p[i] = 0.0f;
}

// ---------------------------------------------------------------- row pass
// R[b][i] = logsumexp_j( A[b][i][j]/lambda - C[b][j] )
// one 256-thread block per row (8 waves): per-thread max over 8 regs,
// single exp per element, then shfl/LDS reduction.
__global__ __launch_bounds__(256) void sk_row_lse(const float* __restrict__ A,
                                                  const float* __restrict__ C,
                                                  float* __restrict__ R) {
    const int row = blockIdx.x;          // 0 .. NB*NN-1
    const int b   = row >> 11;
    const int t   = threadIdx.x;

    const float4* a4 = (const float4*)(A + (size_t)row * NN);
    const float4* c4 = (const float4*)(C + b * NN);

    float4 a0 = a4[t],      c0 = c4[t];
    float4 a1 = a4[t + 256], c1 = c4[t + 256];

    float x0 = a0.x * INV_LAMBDA - c0.x;
    float x1 = a0.y * INV_LAMBDA - c0.y;
    float x2 = a0.z * INV_LAMBDA - c0.z;
    float x3 = a0.w * INV_LAMBDA - c0.w;
    float x4 = a1.x * INV_LAMBDA - c1.x;
    float x5 = a1.y * INV_LAMBDA - c1.y;
    float x6 = a1.z * INV_LAMBDA - c1.z;
    float x7 = a1.w * INV_LAMBDA - c1.w;

    float m = fmaxf(fmaxf(fmaxf(x0, x1), fmaxf(x2, x3)),
                    fmaxf(fmaxf(x4, x5), fmaxf(x6, x7)));
    float s = __expf(x0 - m) + __expf(x1 - m) + __expf(x2 - m) + __expf(x3 - m) +
              __expf(x4 - m) + __expf(x5 - m) + __expf(x6 - m) + __expf(x7 - m);

    // wave32 reduction of (m, s)
#pragma unroll
    for (int off = 16; off > 0; off >>= 1) {
        float mo = __shfl_xor(m, off);
        float so = __shfl_xor(s, off);
        float mn = fmaxf(m, mo);
        s = s * __expf(m - mn) + so * __expf(mo - mn);
        m = mn;
    }

    __shared__ float Ms[8];
    __shared__ float Ss[8];
    if ((t & 31) == 0) { Ms[t >> 5] = m; Ss[t >> 5] = s; }
    __syncthreads();
    if (t == 0) {
        float M = Ms[0], S = Ss[0];
#pragma unroll
        for (int w = 1; w < 8; ++w) {
            float mo = Ms[w], so = Ss[w];
            float mn = fmaxf(M, mo);
            S = S * __expf(M - mn) + so * __expf(mo - mn);
            M = mn;
        }
        R[row] = M + __logf(S);
    }
}

// ---------------------------------------------------------------- column pass
// partial[b][chunk][j] = sum_{i in chunk} exp( A[b][i][j]/lambda - R[b][i] - C[b][j] )
// All exp arguments are <= 0 (R dominates), so no max scan is needed.
// Each wave owns 64 columns: lanes load float4 (global_load_b128, fully
// coalesced), feeding 4 concurrent V_WMMA_F32_16X16X4_F32 column-tiles
// (tile c = columns colbase + 4*lane + c). D = ones(16x4) x B(4x16) + C sums
// over all K slots, so the result is invariant to the exact K <-> (VGPR,
// half-wave) operand mapping. Final write: one coalesced float4 per lane.
__global__ __launch_bounds__(256) void sk_col_wmma(const float* __restrict__ A,
                                                   const float* __restrict__ R,
                                                   const float* __restrict__ Cshift,
                                                   float* __restrict__ partial) {
    const int bid   = blockIdx.x;        // NB * CGROUPS * RCHUNKS blocks
    const int b     = bid >> 6;          // / (CGROUPS*RCHUNKS)
    const int rem   = bid & 63;
    const int cg    = rem >> 4;          // 0..3  (512-column group)
    const int chunk = rem & 15;          // 0..15 (128-row chunk)
    const int t     = threadIdx.x;
    const int w     = t >> 5;            // wave id (8 waves)
    const int lane  = t & 31;
    const int q     = lane & 15;
    const int rp    = lane >> 4;         // row parity within a VGPR half-split
    const int row0  = chunk * CHUNK_ROWS;

    __shared__ float Rsh[CHUNK_ROWS];
    if (t < CHUNK_ROWS) Rsh[t] = R[b * NN + row0 + t];
    __syncthreads();

    const int colbase = cg * 512 + w * 64 + 4 * q;  // this lane's 4 columns
    const float4 sh = *(const float4*)(Cshift + b * NN + colbase);
    const float* Ab = A + (size_t)b * NN * NN;

    v2f aones; aones[0] = 1.0f; aones[1] = 1.0f;
    v8f acc0 = {}, acc1 = {}, acc2 = {}, acc3 = {};

#pragma unroll 2
    for (int k = 0; k < CHUNK_ROWS; k += 4) {
        const int r0 = row0 + k + rp;                 // rows k, k+1
        const float* p0 = Ab + (size_t)r0 * NN + colbase;
        float4 a0 = *(const float4*)p0;               // row r0,   4 cols
        float4 a1 = *(const float4*)(p0 + 2 * NN);    // row r0+2, 4 cols
        float rv0 = Rsh[k + rp];
        float rv1 = Rsh[k + 2 + rp];

        v2f b0, b1, b2, b3;
        b0[0] = __expf(a0.x * INV_LAMBDA - rv0 - sh.x);
        b0[1] = __expf(a1.x * INV_LAMBDA - rv1 - sh.x);
        b1[0] = __expf(a0.y * INV_LAMBDA - rv0 - sh.y);
        b1[1] = __expf(a1.y * INV_LAMBDA - rv1 - sh.y);
        b2[0] = __expf(a0.z * INV_LAMBDA - rv0 - sh.z);
        b2[1] = __expf(a1.z * INV_LAMBDA - rv1 - sh.z);
        b3[0] = __expf(a0.w * INV_LAMBDA - rv0 - sh.w);
        b3[1] = __expf(a1.w * INV_LAMBDA - rv1 - sh.w);

        acc0 = __builtin_amdgcn_wmma_f32_16x16x4_f32(false, aones, false, b0,
                                                     (short)0, acc0, false, false);
        acc1 = __builtin_amdgcn_wmma_f32_16x16x4_f32(false, aones, false, b1,
                                                     (short)0, acc1, false, false);
        acc2 = __builtin_amdgcn_wmma_f32_16x16x4_f32(false, aones, false, b2,
                                                     (short)0, acc2, false, false);
        acc3 = __builtin_amdgcn_wmma_f32_16x16x4_f32(false, aones, false, b3,
                                                     (short)0, acc3, false, false);
    }

    if (lane < 16) {
        float4 o;
        o.x = acc0[0]; o.y = acc1[0]; o.z = acc2[0]; o.w = acc3[0];
        *(float4*)(partial + ((b * RCHUNKS + chunk) * NN) + colbase) = o;
    }
}

// ---------------------------------------------------------------- combine
// C[b][j] = C_old[b][j] + log( sum_chunk partial )   (fixed order -> deterministic)
__global__ __launch_bounds__(256) void sk_col_combine(const float* __restrict__ partial,
                                                      float* __restrict__ C) {
    int j4 = blockIdx.x * 256 + threadIdx.x;   // 0 .. NB*NN/4-1
    int b  = j4 >> 9;                          // 512 float4 per batch row-vector
    int cf = j4 & 511;
    const float4* pp = (const float4*)(partial + (size_t)b * RCHUNKS * NN) + cf;
    float4 s = {0.0f, 0.0f, 0.0f, 0.0f};
#pragma unroll
    for (int c = 0; c < RCHUNKS; ++c) {
        float4 p = pp[c * (NN / 4)];
        s.x += p.x; s.y += p.y; s.z += p.z; s.w += p.w;
    }
    float4* c4 = (float4*)C + j4;
    float4 cv = *c4;
    cv.x += __logf(s.x);
    cv.y += __logf(s.y);
    cv.z += __logf(s.z);
    cv.w += __logf(s.w);
    *c4 = cv;
}

// ---------------------------------------------------------------- final exp
// out = exp( A/lambda - R_i - C_j )
__global__ __launch_bounds__(256) void sk_final_exp(const float* __restrict__ A,
                                                    const float* __restrict__ R,
                                                    const float* __restrict__ C,
                                                    float* __restrict__ out) {
    const size_t total4 = (size_t)NB * NN * (NN / 4);
    const float4* a4 = (const float4*)A;
    float4* o4 = (float4*)out;
    for (size_t i = blockIdx.x * 256ull + threadIdx.x; i < total4;
         i += (size_t)gridDim.x * 256ull) {
        int row = (int)(i >> 9);                       // NN/4 = 512 float4 per row
        int cf  = (int)(i & 511);
        float r = R[row];
        float4 c = ((const float4*)(C + (row >> 11) * NN))[cf];
        float4 a = a4[i];
        float4 o;
        o.x = __expf(a.x * INV_LAMBDA - r - c.x);
        o.y = __expf(a.y * INV_LAMBDA - r - c.y);
        o.z = __expf(a.z * INV_LAMBDA - r - c.z);
        o.w = __expf(a.w * INV_LAMBDA - r - c.w);
        o4[i] = o;
    }
}

// ---------------------------------------------------------------- launcher
extern "C" void kernel_launch(void* const* d_in, const int* in_sizes, int n_in,
                              void* d_out, int out_size, void* d_ws, size_t ws_size,
                              hipStream_t stream) {
    const float* A  = (const float*)d_in[0];
    float* out      = (float*)d_out;

    float* R        = (float*)d_ws;            // NB*NN floats
    float* C        = R + NB * NN;             // NB*NN floats
    float* partial  = C + NB * NN;             // NB*RCHUNKS*NN floats

    sk_init_zero<<<(NB * NN + 255) / 256, 256, 0, stream>>>(C, NB * NN);

    for (int it = 0; it < NIT; ++it) {
        sk_row_lse<<<NB * NN, 256, 0, stream>>>(A, C, R);
        sk_col_wmma<<<NB * CGROUPS * RCHUNKS, 256, 0, stream>>>(A, R, C, partial);
        sk_col_combine<<<(NB * NN / 4) / 256, 256, 0, stream>>>(partial, C);
    }

    sk_final_exp<<<8192, 256, 0, stream>>>(A, R, C, out);
}